// A3C_Network_38903813767521
// MI455X (gfx1250) — compile-verified
//
#include <hip/hip_runtime.h>
#include <math.h>

typedef __attribute__((ext_vector_type(2))) float v2f;
typedef __attribute__((ext_vector_type(8))) float v8f;

__device__ __forceinline__ float lrelu(float v) { return v > 0.0f ? v : 0.1f * v; }
__device__ __forceinline__ float sigmoidf(float x) { return 1.0f / (1.0f + expf(-x)); }

// GEMV tile via V_WMMA_F32_16X16X4_F32: computes 16 output rows
// [row_base, row_base+16) of  W[ldw-major] @ vec, accumulated into acc.
// B is the activation vector broadcast across all 16 N columns, so every
// column of D holds the same result; lane 0 ends with rows m..m+7 in its
// 8 C VGPRs, lane 16 with rows m+8..m+15 (per the ISA C/D layout).
__device__ __forceinline__ v8f gemv_tile(const float* __restrict__ W, int ldw,
                                         int row_base, const float* __restrict__ vec,
                                         int K, int lane, v8f acc)
{
    const int koff = (lane < 16) ? 0 : 2;           // A/B lane-half K offset
    const int m    = row_base + (lane & 15);
    const float* wr = W + m * ldw + koff;
    const float* vr = vec + koff;
    for (int k = 0; k < K; k += 4) {
        v2f a; a.x = wr[k]; a.y = wr[k + 1];        // A: 16x4 f32 layout
        v2f b; b.x = vr[k]; b.y = vr[k + 1];        // B: 4x16, all columns = vec
        acc = __builtin_amdgcn_wmma_f32_16x16x4_f32(
            /*neg_a=*/false, a, /*neg_b=*/false, b,
            /*c_mod=*/(short)0, acc, /*reuse_a=*/false, /*reuse_b=*/false);
    }
    return acc;
}

// Lanes 0 and 16 each own 8 of the 16 tile rows (all columns identical).
__device__ __forceinline__ void store_tile(v8f acc, const float* __restrict__ bias,
                                           float* __restrict__ dst, int row_base,
                                           int lane, bool act)
{
    if ((lane & 15) == 0) {
        const int rb = row_base + ((lane >> 4) << 3);
#pragma unroll
        for (int j = 0; j < 8; ++j) {
            float v = acc[j] + bias[rb + j];
            dst[rb + j] = act ? lrelu(v) : v;
        }
    }
}

__global__ void __launch_bounds__(256, 1)
a3c_fused_kernel(const float* __restrict__ x,   const float* __restrict__ hx,
                 const float* __restrict__ cx,
                 const float* __restrict__ w1,  const float* __restrict__ b1,
                 const float* __restrict__ w2,  const float* __restrict__ b2,
                 const float* __restrict__ w3,  const float* __restrict__ b3,
                 const float* __restrict__ w4,  const float* __restrict__ b4,
                 const float* __restrict__ w_ih, const float* __restrict__ b_ih,
                 const float* __restrict__ w_hh, const float* __restrict__ b_hh,
                 const float* __restrict__ w_cr, const float* __restrict__ b_cr,
                 const float* __restrict__ w_a1, const float* __restrict__ b_a1,
                 const float* __restrict__ w_a2, const float* __restrict__ b_a2,
                 float* __restrict__ out)
{
    __shared__ float s_x[32];
    __shared__ float s_h1[256];
    __shared__ float s_h2[256];
    __shared__ float s_h3[128];
    __shared__ float s_h4[128];
    __shared__ float s_hx[128];
    __shared__ float s_cx[128];
    __shared__ float s_gates[512];
    __shared__ float s_hnew[128];

    const int tid  = threadIdx.x;
    const int lane = tid & 31;
    const int wave = tid >> 5;

    if (tid < 32) s_x[tid] = (tid < 29) ? x[tid] : 0.0f;
    if (tid < 128) { s_hx[tid] = hx[tid]; s_cx[tid] = cx[tid]; }
    __syncthreads();

    // ---- layer 1: [256,29] @ x  (K=29: plain per-thread dot) ----
    {
        float acc = b1[tid];
        const float* wr = w1 + tid * 29;
#pragma unroll
        for (int k = 0; k < 29; ++k) acc += wr[k] * s_x[k];
        s_h1[tid] = lrelu(acc);
    }
    __syncthreads();

    // ---- layer 2: [256,256] @ h1 via WMMA (2 row-tiles per wave) ----
    {
        const v8f z = {};
#pragma unroll
        for (int t = 0; t < 2; ++t) {
            const int rb = (wave * 2 + t) * 16;
            v8f acc = gemv_tile(w2, 256, rb, s_h1, 256, lane, z);
            store_tile(acc, b2, s_h2, rb, lane, true);
        }
    }
    __syncthreads();

    // ---- layer 3: [128,256] @ h2 via WMMA (1 tile per wave) ----
    {
        const v8f z = {};
        const int rb = wave * 16;
        v8f acc = gemv_tile(w3, 256, rb, s_h2, 256, lane, z);
        store_tile(acc, b3, s_h3, rb, lane, true);
    }
    __syncthreads();

    // ---- layer 4: [128,128] @ h3 via WMMA ----
    {
        const v8f z = {};
        const int rb = wave * 16;
        v8f acc = gemv_tile(w4, 128, rb, s_h3, 128, lane, z);
        store_tile(acc, b4, s_h4, rb, lane, true);
    }
    __syncthreads();

    // ---- LSTM gates: [512,128]@h4 + [512,128]@hx, fused accumulator ----
#pragma unroll
    for (int p = 0; p < 4; ++p) {
        const int rb = 16 * (wave + 8 * p);
        v8f acc = {};
        acc = gemv_tile(w_ih, 128, rb, s_h4, 128, lane, acc);
        acc = gemv_tile(w_hh, 128, rb, s_hx, 128, lane, acc);
        if ((lane & 15) == 0) {
            const int r0 = rb + ((lane >> 4) << 3);
#pragma unroll
            for (int j = 0; j < 8; ++j)
                s_gates[r0 + j] = acc[j] + b_ih[r0 + j] + b_hh[r0 + j];
        }
    }
    __syncthreads();

    // ---- LSTM element-wise (gate order i,f,g,o) ----
    if (tid < 128) {
        float ig = sigmoidf(s_gates[tid]);
        float fg = sigmoidf(s_gates[128 + tid]);
        float gg = tanhf(s_gates[256 + tid]);
        float og = sigmoidf(s_gates[384 + tid]);
        float c  = fg * s_cx[tid] + ig * gg;
        s_hnew[tid] = og * tanhf(c);
    }
    __syncthreads();

    // ---- heads: value, softsign(actor), actor2 -> out[0..6] ----
    if (tid < 7) {
        const float* w; float b;
        if (tid == 0)      { w = w_cr;                  b = b_cr[0];       }
        else if (tid < 4)  { w = w_a1 + (tid - 1) * 128; b = b_a1[tid - 1]; }
        else               { w = w_a2 + (tid - 4) * 128; b = b_a2[tid - 4]; }
        float acc = b;
        for (int k = 0; k < 128; ++k) acc += w[k] * s_hnew[k];
        if (tid >= 1 && tid <= 3) acc = acc / (1.0f + fabsf(acc));  // soft_sign
        out[tid] = acc;
    }
}

extern "C" void kernel_launch(void* const* d_in, const int* in_sizes, int n_in,
                              void* d_out, int out_size, void* d_ws, size_t ws_size,
                              hipStream_t stream) {
    (void)in_sizes; (void)n_in; (void)out_size; (void)d_ws; (void)ws_size;
    a3c_fused_kernel<<<1, 256, 0, stream>>>(
        (const float*)d_in[0],  (const float*)d_in[1],  (const float*)d_in[2],
        (const float*)d_in[3],  (const float*)d_in[4],
        (const float*)d_in[5],  (const float*)d_in[6],
        (const float*)d_in[7],  (const float*)d_in[8],
        (const float*)d_in[9],  (const float*)d_in[10],
        (const float*)d_in[11], (const float*)d_in[12],
        (const float*)d_in[13], (const float*)d_in[14],
        (const float*)d_in[15], (const float*)d_in[16],
        (const float*)d_in[17], (const float*)d_in[18],
        (const float*)d_in[19], (const float*)d_in[20],
        (float*)d_out);
}